// CoPooling_78297253806796
// MI455X (gfx1250) — compile-verified
//
#include <hip/hip_runtime.h>
#include <hip/hip_bf16.h>

// ---------------- problem constants ----------------
#define BSZ_    8
#define N_      1024
#define E_      49152
#define D_      256
#define KITER_  10
#define KNODES_ 717
#define MPAD_   720
static const long NN_ = (long)N_ * N_;

typedef __attribute__((ext_vector_type(2))) float v2f;
typedef __attribute__((ext_vector_type(8))) float v8f;

// ---- CDNA5 async global->LDS staging (ISA 15.18.3 op 98, ASYNCcnt) ----
__device__ __forceinline__ void async_b128(float* lds_ptr, const float* gptr) {
    unsigned lds_off = (unsigned)(unsigned long long)lds_ptr;  // LDS aperture: addr[31:0]
    asm volatile("global_load_async_to_lds_b128 %0, %1, off"
                 :: "v"(lds_off), "v"(gptr) : "memory");
}
__device__ __forceinline__ void wait_async0() {
    asm volatile("s_wait_asynccnt 0x0" ::: "memory");
}

// =====================================================================
// WMMA f32 16x16x4 GEMM, 64x64 block per 128-thread (4-wave) group.
// Panels staged in LDS via async copies; A fragment reused across 4 N-tiles.
// MODE_A: 0 = A[m,k], 1 = A[k,m], 2 = A[perm[m],k],
//         3 = concat: k<256 -> A[perm[m],k] else A2[m,k-256]
// TRANS_B: 0 = B[k,n], 1 = B[n,k]
// EPI: 0 = store; 1 = store + acc += gamma[0]*v; 2 = store (v + bias[n])
// Requires: Kk % 32 == 0, grid covers exact 64x64 blocks.
// =====================================================================
template <int MODE_A, int TRANS_B, int EPI>
__global__ __launch_bounds__(128) void wmma_gemm_k(
    const float* __restrict__ A, long sA, int lda,
    const float* __restrict__ B, long sB, int ldb,
    float* __restrict__ C, long sC, int ldc,
    const float* __restrict__ A2, long sA2,
    const int* __restrict__ perm, int sPerm,
    float* __restrict__ acc, long sAcc, const float* __restrict__ gamma,
    const float* __restrict__ bias,
    int Kk, int m_valid, int m_gather)
{
    // padded panels: [64][36] (stride 144B, 16B-aligned rows) or [32][68]
    __shared__ float ldsA[64 * 36];
    __shared__ float ldsB[64 * 36];

    const int b = blockIdx.z;
    A += (long)b * sA;
    B += (long)b * sB;
    C += (long)b * sC;
    if (MODE_A == 3) A2 += (long)b * sA2;
    if (MODE_A == 2 || MODE_A == 3) perm += (long)b * sPerm;
    if (EPI == 1) acc += (long)b * sAcc;

    const int tidx = threadIdx.x;
    const int wave = tidx >> 5;
    const int lane = tidx & 31;
    const int mr   = lane & 15;
    const int kr   = (lane >> 4) << 1;          // 0 or 2
    const int bn   = blockIdx.x * 64;           // N block base
    const int bm   = blockIdx.y * 64;           // M block base
    const int m_local = wave * 16 + mr;         // row within block this lane feeds

    v8f z = {0.f, 0.f, 0.f, 0.f, 0.f, 0.f, 0.f, 0.f};
    v8f cacc[4] = {z, z, z, z};

    for (int k0 = 0; k0 < Kk; k0 += 32) {
        // ---- stage A panel (2048 floats = 512 b128 segs, 4 per thread) ----
#pragma unroll
        for (int s = tidx; s < 512; s += 128) {
            if (MODE_A == 1) {                  // panel layout [k][m] : [32][68]
                int krow = s >> 4, cseg = s & 15;
                async_b128(&ldsA[krow * 68 + cseg * 4],
                           &A[(long)(k0 + krow) * lda + bm + cseg * 4]);
            } else {                            // panel layout [m][k] : [64][36]
                int row = s >> 3, cseg = s & 7;
                const float* g;
                if (MODE_A == 0) {
                    g = &A[(long)(bm + row) * lda + k0 + cseg * 4];
                } else {
                    int mm = bm + row; if (mm > m_gather - 1) mm = m_gather - 1;
                    int gmr = perm[mm];
                    if (MODE_A == 2) {
                        g = &A[(long)gmr * lda + k0 + cseg * 4];
                    } else {                    // MODE_A == 3 (k0 uniform side)
                        if (k0 < 256) {
                            g = &A[(long)gmr * lda + k0 + cseg * 4];
                        } else {
                            int pr = bm + row; if (pr > MPAD_ - 1) pr = MPAD_ - 1;
                            g = &A2[(long)pr * 256 + (k0 - 256) + cseg * 4];
                        }
                    }
                }
                async_b128(&ldsA[row * 36 + cseg * 4], g);
            }
        }
        // ---- stage B panel ----
#pragma unroll
        for (int s = tidx; s < 512; s += 128) {
            if (!TRANS_B) {                     // [k][n] : [32][68]
                int krow = s >> 4, cseg = s & 15;
                async_b128(&ldsB[krow * 68 + cseg * 4],
                           &B[(long)(k0 + krow) * ldb + bn + cseg * 4]);
            } else {                            // [n][k] : [64][36]
                int nrow = s >> 3, cseg = s & 7;
                async_b128(&ldsB[nrow * 36 + cseg * 4],
                           &B[(long)(bn + nrow) * ldb + k0 + cseg * 4]);
            }
        }
        wait_async0();
        __syncthreads();

        // ---- 8 K-steps x 4 N-tiles; A fragment loaded once per K-step ----
#pragma unroll
        for (int ks = 0; ks < 8; ++ks) {
            const int kk = ks * 4 + kr;         // even, 0..30
            v2f a;
            if (MODE_A == 1) {
                a.x = ldsA[kk * 68 + m_local];
                a.y = ldsA[(kk + 1) * 68 + m_local];
            } else {
                const float* p = &ldsA[m_local * 36 + kk];
                a.x = p[0]; a.y = p[1];
            }
#pragma unroll
            for (int nt = 0; nt < 4; ++nt) {
                const int n_local = nt * 16 + mr;
                v2f bf;
                if (!TRANS_B) {
                    bf.x = ldsB[kk * 68 + n_local];
                    bf.y = ldsB[(kk + 1) * 68 + n_local];
                } else {
                    const float* p = &ldsB[n_local * 36 + kk];
                    bf.x = p[0]; bf.y = p[1];
                }
                cacc[nt] = __builtin_amdgcn_wmma_f32_16x16x4_f32(
                               false, a, false, bf, (short)0, cacc[nt],
                               false, false);
            }
        }
        __syncthreads();
    }

    // ---- epilogue: 4 tiles of 16x16 ----
    const int mb0 = bm + wave * 16 + ((lane >> 4) << 3);
    const float g = (EPI == 1) ? gamma[0] : 0.f;
#pragma unroll
    for (int nt = 0; nt < 4; ++nt) {
        const int ncol = bn + nt * 16 + mr;
#pragma unroll
        for (int j = 0; j < 8; ++j) {
            const int mrow = mb0 + j;
            if (mrow < m_valid) {
                const long off = (long)mrow * ldc + ncol;
                float v = cacc[nt][j];
                if (EPI == 0)      C[off] = v;
                else if (EPI == 1) { C[off] = v; acc[off] += g * v; }
                else               C[off] = v + bias[ncol];
            }
        }
    }
}

// ---------------- helper kernels ----------------
__global__ void fill_f32_kernel(float* p, long n, float v) {
    long i = (long)blockIdx.x * blockDim.x + threadIdx.x;
    long st = (long)gridDim.x * blockDim.x;
    for (; i < n; i += st) p[i] = v;
}

__global__ void accum_deg_kernel(const int* __restrict__ tail, float* deg) {
    long i = (long)blockIdx.x * blockDim.x + threadIdx.x;
    if (i >= (long)BSZ_ * E_) return;
    int b = (int)(i / E_);
    atomicAdd(&deg[b * N_ + tail[i]], 1.0f);
}

__global__ void dinv_kernel(const float* __restrict__ deg, float* dinv) {
    int i = blockIdx.x * blockDim.x + threadIdx.x;
    if (i >= BSZ_ * N_) return;
    float d = deg[i];
    dinv[i] = d > 0.f ? rsqrtf(d) : 0.f;
}

__global__ void build_P_edges_kernel(const int* __restrict__ head,
                                     const int* __restrict__ tail,
                                     const float* __restrict__ dinv, float* P) {
    long i = (long)blockIdx.x * blockDim.x + threadIdx.x;
    if (i >= (long)BSZ_ * E_) return;
    int b = (int)(i / E_);
    int h = head[i], t = tail[i];
    float nrm = dinv[b * N_ + h] * dinv[b * N_ + t];
    atomicAdd(&P[(long)b * NN_ + (long)t * N_ + h], nrm);
}

__global__ void build_P_diag_kernel(const float* __restrict__ dinv, float* P) {
    int i = blockIdx.x * blockDim.x + threadIdx.x;
    if (i >= BSZ_ * N_) return;
    int b = i / N_, n = i % N_;
    float dv = dinv[i];
    P[(long)b * NN_ + (long)n * N_ + n] = dv * dv;
}

__global__ void init_acc_kernel(const float* __restrict__ x,
                                const float* __restrict__ temp, float* acc, long n) {
    float t0 = temp[0];
    long i = (long)blockIdx.x * blockDim.x + threadIdx.x;
    long st = (long)gridDim.x * blockDim.x;
    for (; i < n; i += st) acc[i] = t0 * x[i];
}

__global__ void cvec_kernel(const float* __restrict__ Wproj,
                            const float* __restrict__ asrc,
                            const float* __restrict__ atgt,
                            float* csrc, float* ctgt) {
    int k = threadIdx.x;
    float s = 0.f, t = 0.f;
    for (int j = 0; j < D_; ++j) {
        float w = Wproj[j * D_ + k];
        s += asrc[j] * w;
        t += atgt[j] * w;
    }
    csrc[k] = s; ctgt[k] = t;
}

__global__ void svec_kernel(const float* __restrict__ xcut,
                            const float* __restrict__ csrc,
                            const float* __restrict__ ctgt,
                            float* ssrc, float* stgt) {
    int bn = blockIdx.x;
    int t  = threadIdx.x;
    float v = xcut[(long)bn * D_ + t];
    __shared__ float r1[256], r2[256];
    r1[t] = v * csrc[t];
    r2[t] = v * ctgt[t];
    __syncthreads();
    for (int s = 128; s > 0; s >>= 1) {
        if (t < s) { r1[t] += r1[t + s]; r2[t] += r2[t + s]; }
        __syncthreads();
    }
    if (t == 0) { ssrc[bn] = r1[0]; stgt[bn] = r2[0]; }
}

__global__ void scatter_attn_kernel(const int* __restrict__ head,
                                    const int* __restrict__ tail,
                                    const float* __restrict__ ssrc,
                                    const float* __restrict__ stgt,
                                    float* S, float* Cn) {
    long i = (long)blockIdx.x * blockDim.x + threadIdx.x;
    if (i >= (long)BSZ_ * E_) return;
    int b = (int)(i / E_);
    int h = head[i], t = tail[i];
    float z = ssrc[b * N_ + h] + stgt[b * N_ + t];
    float a = 1.f / (1.f + expf(-z));
    float* Sb = S  + (long)b * NN_;
    float* Cb = Cn + (long)b * NN_;
    atomicAdd(&Sb[(long)h * N_ + t], a);
    atomicAdd(&Sb[(long)t * N_ + h], a);
    atomicAdd(&Cb[(long)h * N_ + t], 1.f);
    atomicAdd(&Cb[(long)t * N_ + h], 1.f);
}

__global__ void coalesce_kernel(float* S, const float* __restrict__ Cn, int* Mcount) {
    long i = (long)blockIdx.x * blockDim.x + threadIdx.x;
    int b = (int)(i >> 20);
    float cn = Cn[i];
    float s  = S[i];
    S[i] = (cn > 0.f) ? s / fmaxf(cn, 1.f) : 0.f;
    __shared__ int lc;
    if (threadIdx.x == 0) lc = 0;
    __syncthreads();
    if (cn > 0.f) atomicAdd(&lc, 1);
    __syncthreads();
    if (threadIdx.x == 0 && lc) atomicAdd(&Mcount[b], lc);
}

__global__ void init_select_kernel(const int* __restrict__ Mcount, unsigned* lob,
                                   unsigned* hib, unsigned* cnt, int* krank,
                                   float* fracv) {
    int b = threadIdx.x;
    if (b >= BSZ_) return;
    int M = Mcount[b];
    float pos = 0.2f * (float)(M - 1);
    int lo = (int)floorf(pos);
    float frac = pos - (float)lo;
    int hi = min(lo + 1, M - 1);
    krank[b * 2] = lo; krank[b * 2 + 1] = hi;
    fracv[b] = frac;
    lob[b * 2] = 0u; lob[b * 2 + 1] = 0u;
    hib[b * 2] = 0x3F800000u; hib[b * 2 + 1] = 0x3F800000u;
    cnt[b * 2] = 0u; cnt[b * 2 + 1] = 0u;
}

__global__ void count_le_kernel(const float* __restrict__ A,
                                const unsigned* __restrict__ lob,
                                const unsigned* __restrict__ hib, unsigned* cnt) {
    int b = blockIdx.z;
    const float* Ab = A + (long)b * NN_;
    unsigned l0 = lob[b * 2],     h0 = hib[b * 2];
    unsigned l1 = lob[b * 2 + 1], h1 = hib[b * 2 + 1];
    float m0 = __uint_as_float(l0 + ((h0 - l0) >> 1));
    float m1 = __uint_as_float(l1 + ((h1 - l1) >> 1));
    unsigned c0 = 0, c1 = 0;
    long stride = (long)gridDim.x * blockDim.x;
    for (long i = (long)blockIdx.x * blockDim.x + threadIdx.x; i < NN_; i += stride) {
        float v = Ab[i];
        if (v > 0.f) { c0 += (v <= m0); c1 += (v <= m1); }
    }
    __shared__ unsigned r0[256], r1[256];
    int t = threadIdx.x;
    r0[t] = c0; r1[t] = c1;
    __syncthreads();
    for (int s = 128; s > 0; s >>= 1) {
        if (t < s) { r0[t] += r0[t + s]; r1[t] += r1[t + s]; }
        __syncthreads();
    }
    if (t == 0) {
        atomicAdd(&cnt[b * 2], r0[0]);
        atomicAdd(&cnt[b * 2 + 1], r1[0]);
    }
}

__global__ void update_bounds_kernel(unsigned* lob, unsigned* hib, unsigned* cnt,
                                     const int* __restrict__ krank) {
    int i = threadIdx.x;
    if (i >= 2 * BSZ_) return;
    unsigned l = lob[i], h = hib[i];
    if (l < h) {
        unsigned mid = l + ((h - l) >> 1);
        if (cnt[i] >= (unsigned)(krank[i] + 1)) hib[i] = mid;
        else                                    lob[i] = mid + 1;
    }
    cnt[i] = 0;
}

__global__ void finalize_cut_kernel(const unsigned* __restrict__ lob,
                                    const float* __restrict__ fracv, float* cutv) {
    int b = threadIdx.x;
    if (b >= BSZ_) return;
    float svlo = __uint_as_float(lob[b * 2]);
    float svhi = __uint_as_float(lob[b * 2 + 1]);
    cutv[b] = svlo + fracv[b] * (svhi - svlo);
}

__global__ void threshold_kernel(float* S, const float* __restrict__ cutv) {
    long i = (long)blockIdx.x * blockDim.x + threadIdx.x;
    int b = (int)(i >> 20);
    float v = S[i];
    S[i] = (v >= cutv[b]) ? v : 0.f;
}

__global__ void score_kernel(const float* __restrict__ out, float* score) {
    int bn = blockIdx.x;
    int t  = threadIdx.x;
    __shared__ float r[256];
    r[t] = fabsf(out[(long)bn * D_ + t]);
    __syncthreads();
    for (int s = 128; s > 0; s >>= 1) {
        if (t < s) r[t] += r[t + s];
        __syncthreads();
    }
    if (t == 0) score[bn] = r[0] + 1e-7f;
}

__global__ void topk_kernel(const float* __restrict__ score,
                            const int* __restrict__ labels,
                            const int* __restrict__ ids,
                            int* perm_out, int* labels_out, int* ids_out) {
    int b = blockIdx.x;
    int t = threadIdx.x;
    __shared__ float s[N_];
    __shared__ int   id[N_];
    s[t]  = score[b * N_ + t];
    id[t] = t;
    __syncthreads();
    for (int k = 2; k <= N_; k <<= 1) {
        for (int j = k >> 1; j > 0; j >>= 1) {
            int ixj = t ^ j;
            if (ixj > t) {
                float s1 = s[t], s2 = s[ixj];
                int   i1 = id[t], i2 = id[ixj];
                bool after = (s1 < s2) || (s1 == s2 && i1 > i2);
                bool up = ((t & k) == 0);
                if (up ? after : !after) {
                    s[t] = s2; s[ixj] = s1;
                    id[t] = i2; id[ixj] = i1;
                }
            }
            __syncthreads();
        }
    }
    if (t < KNODES_) {
        int p = id[t];
        perm_out[b * N_ + t]        = p;
        labels_out[b * KNODES_ + t] = labels[b * N_ + p];
        ids_out[b * KNODES_ + t]    = ids[b * N_ + p];
    }
}

// ---------------- host-side orchestration ----------------
extern "C" void kernel_launch(void* const* d_in, const int* in_sizes, int n_in,
                              void* d_out, int out_size, void* d_ws, size_t ws_size,
                              hipStream_t stream) {
    const float* x      = (const float*)d_in[0];
    const int*   head   = (const int*)d_in[2];
    const int*   tail   = (const int*)d_in[3];
    const int*   labels = (const int*)d_in[5];
    const int*   ids    = (const int*)d_in[6];
    const float* temp   = (const float*)d_in[7];
    const float* Wproj  = (const float*)d_in[8];
    const float* asrc   = (const float*)d_in[9];
    const float* atgt   = (const float*)d_in[10];
    const float* Wlin   = (const float*)d_in[11];
    const float* blin   = (const float*)d_in[12];

    float* W    = (float*)d_ws;
    float* P    = W;
    float* S    = P + (long)BSZ_ * NN_;
    float* Y0   = S + (long)BSZ_ * NN_;
    float* Y1   = Y0 + (long)BSZ_ * N_ * D_;
    float* ACC  = Y1 + (long)BSZ_ * N_ * D_;
    float* PO   = ACC + (long)BSZ_ * N_ * D_;
    float* deg  = PO + (long)BSZ_ * MPAD_ * D_;
    float* dinv = deg + BSZ_ * N_;
    float* csrc = dinv + BSZ_ * N_;
    float* ctgt = csrc + D_;
    float* ssrc = ctgt + D_;
    float* stgt = ssrc + BSZ_ * N_;
    float* scor = stgt + BSZ_ * N_;
    float* cutv = scor + BSZ_ * N_;
    float* frcv = cutv + BSZ_;
    unsigned* lob = (unsigned*)(frcv + BSZ_);
    unsigned* hib = lob + 2 * BSZ_;
    unsigned* cnt = hib + 2 * BSZ_;
    int* krank  = (int*)(cnt + 2 * BSZ_);
    int* Mcount = krank + 2 * BSZ_;
    int* perm   = Mcount + BSZ_;

    const long nXD = (long)BSZ_ * N_ * D_;
    const long nNN = (long)BSZ_ * NN_;
    const long nBE = (long)BSZ_ * E_;

    // ---- PageRank transition matrix P ----
    fill_f32_kernel<<<4096, 256, 0, stream>>>(P, nNN, 0.f);
    fill_f32_kernel<<<32, 256, 0, stream>>>(deg, BSZ_ * N_, 1.0f);
    accum_deg_kernel<<<(int)((nBE + 255) / 256), 256, 0, stream>>>(tail, deg);
    dinv_kernel<<<(BSZ_ * N_) / 256, 256, 0, stream>>>(deg, dinv);
    build_P_edges_kernel<<<(int)((nBE + 255) / 256), 256, 0, stream>>>(head, tail, dinv, P);
    build_P_diag_kernel<<<(BSZ_ * N_) / 256, 256, 0, stream>>>(dinv, P);

    // ---- PageRank: acc = temp0*x + sum_k temp_k * (P^k x) ----
    init_acc_kernel<<<4096, 256, 0, stream>>>(x, temp, ACC, nXD);
    {
        const float* xin = x;
        float* bufs[2] = {Y0, Y1};
        for (int k = 1; k <= KITER_; ++k) {
            float* yout = bufs[(k - 1) & 1];
            dim3 grid(D_ / 64, N_ / 64, BSZ_);           // 4 x 16 x 8
            wmma_gemm_k<0, 0, 1><<<grid, 128, 0, stream>>>(
                P, NN_, N_, xin, (long)N_ * D_, D_,
                yout, (long)N_ * D_, D_,
                nullptr, 0, nullptr, 0,
                ACC, (long)N_ * D_, temp + k, nullptr,
                N_, N_, 0);
            xin = yout;
        }
    }

    // ---- GAT edge scores (projection folded into D-vectors) ----
    cvec_kernel<<<1, 256, 0, stream>>>(Wproj, asrc, atgt, csrc, ctgt);
    svec_kernel<<<BSZ_ * N_, 256, 0, stream>>>(ACC, csrc, ctgt, ssrc, stgt);

    // ---- symmetrize + coalesce(mean); Cn reuses the P buffer ----
    fill_f32_kernel<<<4096, 256, 0, stream>>>(S, nNN, 0.f);
    fill_f32_kernel<<<4096, 256, 0, stream>>>(P, nNN, 0.f);
    fill_f32_kernel<<<1, 64, 0, stream>>>((float*)Mcount, BSZ_, 0.f);
    scatter_attn_kernel<<<(int)((nBE + 255) / 256), 256, 0, stream>>>(
        head, tail, ssrc, stgt, S, P);
    coalesce_kernel<<<(int)(nNN / 256), 256, 0, stream>>>(S, P, Mcount);

    // ---- exact 20th percentile via bitwise bisection on positive floats ----
    init_select_kernel<<<1, 32, 0, stream>>>(Mcount, lob, hib, cnt, krank, frcv);
    for (int it = 0; it < 31; ++it) {
        dim3 grid(1024, 1, BSZ_);
        count_le_kernel<<<grid, 256, 0, stream>>>(S, lob, hib, cnt);
        update_bounds_kernel<<<1, 32, 0, stream>>>(lob, hib, cnt, krank);
    }
    finalize_cut_kernel<<<1, 32, 0, stream>>>(lob, frcv, cutv);
    threshold_kernel<<<(int)(nNN / 256), 256, 0, stream>>>(S, cutv);

    // ---- out = A_cut^T @ x ; node scores ; top-717 ----
    {
        dim3 grid(D_ / 64, N_ / 64, BSZ_);
        wmma_gemm_k<1, 0, 0><<<grid, 128, 0, stream>>>(
            S, NN_, N_, x, (long)N_ * D_, D_,
            Y0, (long)N_ * D_, D_,
            nullptr, 0, nullptr, 0, nullptr, 0, nullptr, nullptr,
            N_, N_, 0);
    }
    score_kernel<<<BSZ_ * N_, 256, 0, stream>>>(Y0, scor);

    float* hnew_out   = (float*)d_out;
    int*   labels_out = (int*)d_out + (long)BSZ_ * KNODES_ * D_;
    int*   ids_out    = labels_out + BSZ_ * KNODES_;
    topk_kernel<<<BSZ_, N_, 0, stream>>>(scor, labels, ids, perm, labels_out, ids_out);

    // ---- pooled = A_cut[perm] @ x  (M padded to 768, stores guarded) ----
    {
        dim3 grid(D_ / 64, 12, BSZ_);
        wmma_gemm_k<2, 0, 0><<<grid, 128, 0, stream>>>(
            S, NN_, N_, x, (long)N_ * D_, D_,
            PO, (long)MPAD_ * D_, D_,
            nullptr, 0, perm, N_,
            nullptr, 0, nullptr, nullptr,
            N_, MPAD_, KNODES_);
    }

    // ---- h_new = [x[perm] | pooled] @ W_lin^T + b_lin ----
    {
        dim3 grid(D_ / 64, 12, BSZ_);
        wmma_gemm_k<3, 1, 2><<<grid, 128, 0, stream>>>(
            x, (long)N_ * D_, D_,
            Wlin, 0, 2 * D_,
            hnew_out, (long)KNODES_ * D_, D_,
            PO, (long)MPAD_ * D_,
            perm, N_,
            nullptr, 0, nullptr, blin,
            2 * D_, KNODES_, KNODES_);
    }
}